// LocalFeatureNet_54735063220238
// MI455X (gfx1250) — compile-verified
//
#include <hip/hip_runtime.h>

// Problem constants (from reference): image (4,1,32,512,512) f32,
// coords (4,512,3) f32, object_ids (4,512) i32, out (4,512,5,32,32) f32.
#define N_  4
#define S_  512
#define D_  32
#define H_  512
#define W_  512
#define PZ  5
#define P_  32
#define WAVES_PER_BLOCK 8

__global__ __launch_bounds__(256)
void LocalFeatureNet_patch_gather(const float* __restrict__ image,
                                  const float* __restrict__ coords,
                                  const int*   __restrict__ object_ids,
                                  float*       __restrict__ out)
{
    const int lane = threadIdx.x & 31;           // x within patch row (wave32)
    const int wave = threadIdx.x >> 5;

    // One wave handles one (n, s, z) slab of 32x32 output elements.
    int slab = blockIdx.x * WAVES_PER_BLOCK + wave;
    slab = __builtin_amdgcn_readfirstlane(slab); // wave-uniform -> scalar loads

    const int z  = slab % PZ;
    const int ns = slab / PZ;                    // n*S_ + s
    const int n  = ns / S_;

    // z-offset table [0, +1, -1, +2, -2] computed arithmetically (no LDS/scratch)
    const int zoff = ((z + 1) >> 1) * ((z & 1) ? 1 : -1);

    // Wave-uniform per-sample metadata (scalar loads via readfirstlane'd addr)
    const float* c  = coords + (size_t)ns * 3;
    const int pz    = __float2int_rn(c[0]);      // round-to-nearest-even == jnp.round
    const int py    = __float2int_rn(c[1]);
    const int px    = __float2int_rn(c[2]);
    const bool obj_ok = (object_ids[ns] != -1);

    const int zz  = pz + zoff;
    const bool vz = (zz >= 0) && (zz < D_) && obj_ok;
    const int zc  = min(max(zz, 0), D_ - 1);

    const int xx  = px + (lane - P_ / 2);
    const bool vx = (xx >= 0) && (xx < W_);
    const int xc  = min(max(xx, 0), W_ - 1);

    const float* img_slice = image + ((size_t)n * D_ + (size_t)zc) * (size_t)(H_ * W_);
    float* orow = out + (((size_t)ns * PZ + (size_t)z) * (size_t)(P_ * P_)) + lane;

    // Branchless: always load at clamped (in-bounds) address, select 0 for
    // invalid elements. Full unroll -> 32 independent 128B coalesced loads in
    // flight per wave; stores are streaming (non-temporal).
#pragma unroll
    for (int y = 0; y < P_; ++y) {
        const int yy  = py + (y - P_ / 2);
        const bool vy = (yy >= 0) && (yy < H_);
        const int yc  = min(max(yy, 0), H_ - 1);

        float v = img_slice[(size_t)yc * W_ + xc];
        v = (vz && vy && vx) ? v : 0.0f;

        __builtin_nontemporal_store(v, orow + (size_t)y * P_);
    }
}

extern "C" void kernel_launch(void* const* d_in, const int* in_sizes, int n_in,
                              void* d_out, int out_size, void* d_ws, size_t ws_size,
                              hipStream_t stream)
{
    const float* image      = (const float*)d_in[0];
    const float* coords     = (const float*)d_in[1];
    const int*   object_ids = (const int*)  d_in[2];
    float*       out        = (float*)d_out;

    const int slabs  = N_ * S_ * PZ;                 // 10240 waves of work
    const int blocks = slabs / WAVES_PER_BLOCK;      // 1280 blocks x 256 threads

    LocalFeatureNet_patch_gather<<<blocks, 256, 0, stream>>>(image, coords,
                                                            object_ids, out);
}